// PrivacyAwareTokenPruning_4088808866130
// MI455X (gfx1250) — compile-verified
//
#include <hip/hip_runtime.h>
#include <hip/hip_bf16.h>

#define B_DIM      32
#define N_DIM      4096
#define D_DIM      256
#define K_SEL      2048          // N * (1 - PRUNE_RATIO)
#define NOISE_SCALE 0.1f
#define MIXUP_ALPHA 0.05f

typedef __attribute__((ext_vector_type(2))) float v2f;
typedef __attribute__((ext_vector_type(8))) float v8f;

// Monotonic float -> uint mapping (total order incl. negatives).
__device__ __forceinline__ unsigned sortable_f32(float f) {
  unsigned b = __float_as_uint(f);
  return (b & 0x80000000u) ? ~b : (b | 0x80000000u);
}

__device__ __forceinline__ unsigned long long make_key(const float* attn,
                                                       const float* noise,
                                                       int base, int n) {
  float u = noise[base + n];
  float g = -logf(-logf(u + 1e-10f) + 1e-10f);
  float s = attn[base + n] + NOISE_SCALE * g;
  unsigned hi = ~sortable_f32(s);     // ascending key == descending score
  return ((unsigned long long)hi << 32) | (unsigned)n; // ties: lower idx first
}

// ---------------------------------------------------------------------------
// Kernel 1: per-batch sort + pruned-row mean (WMMA).  32 blocks x 1024 thr.
// Writes: wsIdx[b*K_SEL + j] (selected indices, rank order)
//         wsMean[b*D_DIM + d] (mean of pruned rows)
// ---------------------------------------------------------------------------
__global__ __launch_bounds__(1024)
void sort_mean_kernel(const float* __restrict__ seq,
                      const float* __restrict__ attn,
                      const float* __restrict__ noise,
                      int* __restrict__ wsIdx,
                      float* __restrict__ wsMean) {
  __shared__ unsigned long long keys[N_DIM];            // 32 KB
  __shared__ float partial[2 * D_DIM];                  // 2 KB

  const int b   = blockIdx.x;
  const int tid = threadIdx.x;
  const float* __restrict__ seqB = seq + (size_t)b * N_DIM * D_DIM;

  // Phase 1: gumbel-perturbed scores -> sortable keys (softmax skipped:
  // monotonic, preserves top-k values' order and tie-breaks).
  for (int n = tid; n < N_DIM; n += 1024)
    keys[n] = make_key(attn, noise, b * N_DIM, n);
  __syncthreads();

  // Phase 2: ascending bitonic sort (largest score ends at position 0).
  for (unsigned k = 2; k <= N_DIM; k <<= 1) {
    for (unsigned j = k >> 1; j > 0; j >>= 1) {
      for (unsigned i = tid; i < N_DIM; i += 1024) {
        unsigned ixj = i ^ j;
        if (ixj > i) {
          unsigned long long a = keys[i], c = keys[ixj];
          bool up = ((i & k) == 0);
          if ((a > c) == up) { keys[i] = c; keys[ixj] = a; }
        }
      }
      __syncthreads();
    }
  }
  // keys[0..K_SEL-1] = selected (descending), keys[K_SEL..N-1] = pruned.

  // Phase 3: column sums of the 2048 pruned rows via V_WMMA_F32_16X16X4_F32.
  // D = A(16x4, all-ones) x B(4 pruned rows x 16 features) + C: every row of
  // D equals the column sums (invariant to the K-slot lane layout of B).
  // 32 waves = 16 feature tiles x 2 row-halves.
  {
    const int wave = tid >> 5;
    const int lane = tid & 31;
    const int t    = wave >> 1;          // feature tile 0..15
    const int half = wave & 1;           // row half 0..1 (1024 rows each)
    const int d0   = t * 16;
    const int kl   = (lane >> 4) << 1;   // half-wave owns K=0,1 or K=2,3
    const int nsub = lane & 15;
    v2f a;  a.x = 1.0f; a.y = 1.0f;
    v8f acc = {};
    const int j_beg = K_SEL + half * 1024;
    const int j_end = j_beg + 1024;
    for (int j0 = j_beg; j0 < j_end; j0 += 4) {
      unsigned r0 = (unsigned)(keys[j0 + kl]     & 0xFFFFFFFFull);
      unsigned r1 = (unsigned)(keys[j0 + kl + 1] & 0xFFFFFFFFull);
      v2f bv;
      bv.x = seqB[(size_t)r0 * D_DIM + d0 + nsub];
      bv.y = seqB[(size_t)r1 * D_DIM + d0 + nsub];
      acc = __builtin_amdgcn_wmma_f32_16x16x4_f32(
                false, a, false, bv, (short)0, acc, false, false);
    }
    if (lane < 16)                       // D row 0: N = lane
      partial[half * D_DIM + d0 + lane] = acc[0];
  }
  __syncthreads();

  // Phase 4: reduce halves, emit mean + selected indices.
  if (tid < D_DIM)
    wsMean[b * D_DIM + tid] =
        (partial[tid] + partial[D_DIM + tid]) * (1.0f / (float)K_SEL);
  for (int j = tid; j < K_SEL; j += 1024)
    wsIdx[b * K_SEL + j] = (int)(unsigned)(keys[j] & 0xFFFFFFFFull);
}

// ---------------------------------------------------------------------------
// Kernel 2: bandwidth-bound gather + mixup.  4096 blocks x 256 threads,
// 16 output rows per block -> enough concurrency to saturate 23.3 TB/s HBM.
// ---------------------------------------------------------------------------
#define ROWS_PER_BLK 16
__global__ __launch_bounds__(256)
void gather_kernel(const float* __restrict__ seq,
                   const int* __restrict__ wsIdx,
                   const float* __restrict__ wsMean,
                   float* __restrict__ out) {
  __shared__ __align__(16) float meanv[D_DIM];
  __shared__ int rowidx[ROWS_PER_BLK];

  const int tid = threadIdx.x;
  const int b   = blockIdx.x >> 7;                 // 128 blocks per batch
  const int rb  = (blockIdx.x & 127) * ROWS_PER_BLK;

  meanv[tid] = wsMean[b * D_DIM + tid];
  if (tid < ROWS_PER_BLK)
    rowidx[tid] = wsIdx[b * K_SEL + rb + tid];
  __syncthreads();

  const float4* __restrict__ seq4 =
      reinterpret_cast<const float4*>(seq + (size_t)b * N_DIM * D_DIM);
  const float4* __restrict__ mean4 = reinterpret_cast<const float4*>(meanv);
  float4* __restrict__ out4 = reinterpret_cast<float4*>(out) +
      ((size_t)b * K_SEL + rb) * (D_DIM / 4);

  for (int e = tid; e < ROWS_PER_BLK * (D_DIM / 4); e += 256) {
    int r  = e >> 6;                 // local row
    int d4 = e & 63;                 // float4 column
    float4 v = seq4[(size_t)rowidx[r] * (D_DIM / 4) + d4];
    float4 m = mean4[d4];
    float4 o;
    o.x = fmaf(MIXUP_ALPHA, m.x, v.x);
    o.y = fmaf(MIXUP_ALPHA, m.y, v.y);
    o.z = fmaf(MIXUP_ALPHA, m.z, v.z);
    o.w = fmaf(MIXUP_ALPHA, m.w, v.w);
    out4[e] = o;
  }
}

// ---------------------------------------------------------------------------
// Fallback: fused single kernel (used only if workspace is too small).
// ---------------------------------------------------------------------------
__global__ __launch_bounds__(256)
void fused_kernel(const float* __restrict__ seq,
                  const float* __restrict__ attn,
                  const float* __restrict__ noise,
                  float* __restrict__ out) {
  __shared__ unsigned long long keys[N_DIM];
  __shared__ __align__(16) float meanv[D_DIM];

  const int b   = blockIdx.x;
  const int tid = threadIdx.x;
  const float* __restrict__ seqB = seq + (size_t)b * N_DIM * D_DIM;

  for (int n = tid; n < N_DIM; n += 256)
    keys[n] = make_key(attn, noise, b * N_DIM, n);
  __syncthreads();

  for (unsigned k = 2; k <= N_DIM; k <<= 1) {
    for (unsigned j = k >> 1; j > 0; j >>= 1) {
      for (unsigned i = tid; i < N_DIM; i += 256) {
        unsigned ixj = i ^ j;
        if (ixj > i) {
          unsigned long long a = keys[i], c = keys[ixj];
          bool up = ((i & k) == 0);
          if ((a > c) == up) { keys[i] = c; keys[ixj] = a; }
        }
      }
      __syncthreads();
    }
  }

  {
    const int wave = tid >> 5;
    const int lane = tid & 31;
    const int d0   = wave * 32;
    const int kl   = (lane >> 4) << 1;
    const int nsub = lane & 15;
    v2f a;  a.x = 1.0f; a.y = 1.0f;
    v8f acc0 = {}; v8f acc1 = {};
    for (int j0 = K_SEL; j0 < N_DIM; j0 += 4) {
      unsigned r0 = (unsigned)(keys[j0 + kl]     & 0xFFFFFFFFull);
      unsigned r1 = (unsigned)(keys[j0 + kl + 1] & 0xFFFFFFFFull);
      const float* p0 = seqB + (size_t)r0 * D_DIM;
      const float* p1 = seqB + (size_t)r1 * D_DIM;
      v2f b0; b0.x = p0[d0 + nsub];      b0.y = p1[d0 + nsub];
      v2f b1; b1.x = p0[d0 + 16 + nsub]; b1.y = p1[d0 + 16 + nsub];
      acc0 = __builtin_amdgcn_wmma_f32_16x16x4_f32(
                 false, a, false, b0, (short)0, acc0, false, false);
      acc1 = __builtin_amdgcn_wmma_f32_16x16x4_f32(
                 false, a, false, b1, (short)0, acc1, false, false);
    }
    if (lane < 16) {
      meanv[d0 + lane]      = acc0[0] * (1.0f / (float)K_SEL);
      meanv[d0 + 16 + lane] = acc1[0] * (1.0f / (float)K_SEL);
    }
  }
  __syncthreads();

  const float4* __restrict__ seq4  = reinterpret_cast<const float4*>(seqB);
  const float4* __restrict__ mean4 = reinterpret_cast<const float4*>(meanv);
  float4* __restrict__ out4 =
      reinterpret_cast<float4*>(out) + (size_t)b * K_SEL * (D_DIM / 4);
  for (int e = tid; e < K_SEL * (D_DIM / 4); e += 256) {
    int j  = e >> 6;
    int d4 = e & 63;
    unsigned r = (unsigned)(keys[j] & 0xFFFFFFFFull);
    float4 v = seq4[(size_t)r * (D_DIM / 4) + d4];
    float4 m = mean4[d4];
    float4 o;
    o.x = fmaf(MIXUP_ALPHA, m.x, v.x);
    o.y = fmaf(MIXUP_ALPHA, m.y, v.y);
    o.z = fmaf(MIXUP_ALPHA, m.z, v.z);
    o.w = fmaf(MIXUP_ALPHA, m.w, v.w);
    out4[e] = o;
  }
}

extern "C" void kernel_launch(void* const* d_in, const int* in_sizes, int n_in,
                              void* d_out, int out_size, void* d_ws, size_t ws_size,
                              hipStream_t stream) {
  const float* seq   = (const float*)d_in[0];  // (B, N, D) f32
  const float* attn  = (const float*)d_in[1];  // (B, N)    f32
  const float* noise = (const float*)d_in[2];  // (B, N)    f32
  float*       out   = (float*)d_out;          // (B, K, D) f32
  (void)in_sizes; (void)n_in; (void)out_size;

  const size_t meanBytes = (size_t)B_DIM * D_DIM * sizeof(float);   // 32 KB
  const size_t idxBytes  = (size_t)B_DIM * K_SEL * sizeof(int);     // 256 KB

  if (d_ws != nullptr && ws_size >= meanBytes + idxBytes) {
    float* wsMean = (float*)d_ws;
    int*   wsIdx  = (int*)((char*)d_ws + meanBytes);
    sort_mean_kernel<<<B_DIM, 1024, 0, stream>>>(seq, attn, noise,
                                                 wsIdx, wsMean);
    gather_kernel<<<B_DIM * (K_SEL / ROWS_PER_BLK), 256, 0, stream>>>(
        seq, wsIdx, wsMean, out);
  } else {
    fused_kernel<<<B_DIM, 256, 0, stream>>>(seq, attn, noise, out);
  }
}